// IndNonMaximumSuppression_16484084482950
// MI455X (gfx1250) — compile-verified
//
#include <hip/hip_runtime.h>
#include <hip/hip_bf16.h>

typedef __attribute__((ext_vector_type(2))) float v2f;
typedef __attribute__((ext_vector_type(8))) float v8f;

typedef __attribute__((address_space(1))) int gas_int;   // global (AS1)
typedef __attribute__((address_space(3))) int las_int;   // LDS (AS3)

#define NMS_B  16
#define NMS_N  2048
#define NMS_R  256
#define NMS_NT 128   // number of 16-wide tiles along j
#define NMS_NW 64    // 2048 bits of "removed" flags = 64 u32 words

#if __has_builtin(__builtin_amdgcn_global_load_async_to_lds_b32) && \
    __has_builtin(__builtin_amdgcn_s_wait_asynccnt)
#define NMS_ASYNC 1
#else
#define NMS_ASYNC 0
#endif

// One workgroup per batch. 256 threads = 8 waves (wave32).
__global__ __launch_bounds__(256) void nms_fused_kernel(const float* __restrict__ pred,
                                                        int* __restrict__ out)
{
    // 8 KB: bitonic sort keys, later reused as rowbits32[2][8][128] ballot words
    __shared__ float    s_keybits[NMS_N];
    __shared__ int      s_order[NMS_N];                          // 8 KB
    __shared__ float    s_x1[NMS_N], s_y1[NMS_N];                // 16 KB
    __shared__ float    s_x2[NMS_N], s_y2[NMS_N];                // 16 KB
    __shared__ float    s_area[NMS_N];                           // 8 KB
    __shared__ unsigned s_removed[NMS_NW];                       // 256 B
    __shared__ int      s_sel[NMS_R];                            // 1 KB
    __shared__ int      s_cnt;
    __shared__ int      s_done;

    const int b    = blockIdx.x;
    const int tid  = threadIdx.x;
    const int lane = tid & 31;
    const int wave = tid >> 5;
    const int half = (lane >> 4) & 1;   // which 16-lane half of the wave
    const int l16  = lane & 15;
    const float* bp = pred + (size_t)b * NMS_N * 5;

    // ---------------- 1. load sort keys (async global -> LDS where available) ----------------
#if NMS_ASYNC
    for (int i = tid; i < NMS_N; i += 256) {
        __builtin_amdgcn_global_load_async_to_lds_b32(
            (gas_int*)(unsigned long long)(const void*)(bp + (size_t)i * 5),
            (las_int*)(unsigned)(unsigned long long)(void*)&s_keybits[i],
            0, 0);
        s_order[i] = i;
    }
    __builtin_amdgcn_s_wait_asynccnt(0);
#else
    for (int i = tid; i < NMS_N; i += 256) {
        s_keybits[i] = bp[i * 5];       // score
        s_order[i]   = i;
    }
#endif
    __syncthreads();

    // ---------------- 2. bitonic sort: score desc, idx asc on ties ----------------
    for (int k = 2; k <= NMS_N; k <<= 1) {
        for (int j = k >> 1; j > 0; j >>= 1) {
            for (int t = tid; t < NMS_N; t += 256) {
                int p = t ^ j;
                if (p > t) {
                    float ka = s_keybits[t], kb = s_keybits[p];
                    int   ia = s_order[t],   ib = s_order[p];
                    bool before = (ka > kb) || (ka == kb && ia < ib);
                    bool up = ((t & k) == 0);
                    if (up ? !before : before) {
                        s_keybits[t] = kb; s_keybits[p] = ka;
                        s_order[t]   = ib; s_order[p]   = ia;
                    }
                }
            }
            __syncthreads();
        }
    }

    // ---------------- 3. gather corners / areas in sorted order ----------------
    for (int i = tid; i < NMS_N; i += 256) {
        const float* p = bp + (size_t)s_order[i] * 5;
        float x = p[1], y = p[2], w = p[3], h = p[4];
        float ws = floorf(w * 0.5f), hs = floorf(h * 0.5f);
        float x1 = x - ws, y1 = y - hs, x2 = x + ws, y2 = y + hs;
        s_x1[i] = x1; s_y1[i] = y1; s_x2[i] = x2; s_y2[i] = y2;
        s_area[i] = (x2 - x1) * (y2 - y1);   // > 0 always (w,h >= 16)
    }
    if (tid < NMS_NW) s_removed[tid] = 0u;
    if (tid == 0) { s_cnt = 0; s_done = 0; }
    __syncthreads();

    unsigned*          rowbits32 = (unsigned*)s_keybits;   // [2][8][NMS_NT] ballot words
    volatile unsigned* rem       = s_removed;
    volatile int*      pcnt      = &s_cnt;
    volatile int*      pdone     = &s_done;

    // ---- mask builder: 16 rows (block I) vs all j-tiles >= tlo, per wave ----
    // V_WMMA_F32_16X16X4_F32 forms the 16x16 outer sum (area_i + area_j) -- the
    // union term of IoU -- directly in the f32 accumulator layout.
    auto fill_block = [&](int I, int buf, int tlo, int tstep) {
        const int ibase = I << 4;
        float rx1[8], ry1[8], rx2[8], ry2[8];
#pragma unroll
        for (int v = 0; v < 8; ++v) {                 // row m = v + 8*half (C/D layout)
            int m = ibase + v + 8 * half;
            rx1[v] = s_x1[m]; ry1[v] = s_y1[m];
            rx2[v] = s_x2[m]; ry2[v] = s_y2[m];
        }
        // A (16x4 f32): col0 = area_row, col1 = 1. Lanes 0-15 hold K=0,1; lanes 16-31 hold K=2,3 (=0).
        v2f a;
        a.x = half ? 0.0f : s_area[ibase + l16];
        a.y = half ? 0.0f : 1.0f;
        unsigned* dst32 = rowbits32 + buf * (8 * NMS_NT);
        for (int t = tlo; t < NMS_NT; t += tstep) {
            const int jbase = t << 4;
            float cx1 = s_x1[jbase + l16], cy1 = s_y1[jbase + l16];
            float cx2 = s_x2[jbase + l16], cy2 = s_y2[jbase + l16];
            // B (4x16 f32): row0 = 1, row1 = area_col. Lanes 0-15 hold K=0,1; lanes 16-31 K=2,3 (=0).
            v2f bm;
            bm.x = half ? 0.0f : 1.0f;
            bm.y = half ? 0.0f : s_area[jbase + l16];
            v8f c = {};
            // D[m][n] = area_row[m] + area_col[n]
            v8f s = __builtin_amdgcn_wmma_f32_16x16x4_f32(
                false, a, false, bm, (short)0, c, false, false);
            unsigned selm = 0u;
#pragma unroll
            for (int v = 0; v < 8; ++v) {
                int gi = ibase + v + 8 * half;        // global sorted row
                int gj = jbase + l16;                 // global sorted col
                float iw = fminf(rx2[v], cx2) - fmaxf(rx1[v], cx1);
                float ih = fminf(ry2[v], cy2) - fmaxf(ry1[v], cy1);
                iw = fmaxf(iw, 0.0f); ih = fmaxf(ih, 0.0f);
                float inter = iw * ih;
                float uni   = s[v] - inter;
                // union > 0 always; iou > 0.5  <=>  2*inter > union (exact scale by 2)
                bool sup = ((inter + inter) > uni) && (gj > gi);
                unsigned mk = (unsigned)__ballot(sup);   // low16: row v, high16: row v+8
                selm = (lane == v) ? mk : selm;          // branch-free select, mk is uniform
            }
            if (lane < 8) dst32[lane * NMS_NT + t] = selm;  // one store per tile
        }
    };

    // ---- sequential scan of one 16-row block, executed by wave 0 only ----
    auto scan_block = [&](int I, int buf) {
        const unsigned* src32 = (const unsigned*)rowbits32 + buf * (8 * NMS_NT);
        const int wlo = I >> 1;                       // first removed-word containing j >= 16*I
        for (int r = 0; r < 16; ++r) {
            const int gi = (I << 4) + r;
            unsigned kw = rem[gi >> 5];               // LDS broadcast read
            bool kept = ((kw >> (gi & 31)) & 1u) == 0u;
            if (kept) {
                if (lane == 0) {
                    int c = *pcnt;
                    if (c < NMS_R) s_sel[c] = gi;
                    *pcnt = c + 1;
                    if (c + 1 >= NMS_R) *pdone = 1;   // first R kept found: exact early exit
                }
                const int rr = r & 7;
                const int sh = (r >> 3) << 4;         // 0 (rows 0-7) or 16 (rows 8-15)
#pragma unroll
                for (int q = 0; q < 2; ++q) {         // each lane owns 2 of 64 words
                    int w = lane + q * 32;
                    if (w >= wlo) {
                        int t0 = 2 * w, t1 = 2 * w + 1;
                        unsigned m0 = (t0 >= I) ? src32[rr * NMS_NT + t0] : 0u;
                        unsigned m1 = src32[rr * NMS_NT + t1];
                        unsigned bits = ((m0 >> sh) & 0xFFFFu) | (((m1 >> sh) & 0xFFFFu) << 16);
                        if (bits) rem[w] |= bits;
                    }
                }
            }
            __builtin_amdgcn_wave_barrier();          // wave-lockstep; per-wave LDS is in-order
            if (*pdone) break;
        }
    };

    // ---------------- 4. pipelined mask-build + scan ----------------
    fill_block(0, 0, wave, 8);                        // prologue: all 8 waves build block 0
    __syncthreads();
    for (int I = 0; I < NMS_NT; ++I) {
        if (wave == 0)
            scan_block(I, I & 1);                     // sequential part: one wave
        else if (I + 1 < NMS_NT)
            fill_block(I + 1, (I & 1) ^ 1, (I + 1) + (wave - 1), 7);  // overlap next block
        __syncthreads();
        if (*pdone) break;                            // uniform across workgroup
    }
    __syncthreads();

    // ---------------- 5. emit: map kept sorted positions back through order ----------------
    int cnt = *pcnt; if (cnt > NMS_R) cnt = NMS_R;
    for (int r = tid; r < NMS_R; r += 256) {
        int val;
        if (r < cnt) {
            val = s_order[s_sel[r]];
        } else {
            // reference pads invalid slots with random idx in [0, N); emit a hash-based stand-in
            unsigned h = ((unsigned)b * 1103515245u) ^ ((unsigned)r * 2654435769u) ^ 0x9E3779B9u;
            h ^= h >> 16; h *= 2246822519u; h ^= h >> 13;
            val = (int)(h & (NMS_N - 1));
        }
        out[(size_t)b * NMS_R + r] = val;
    }
}

extern "C" void kernel_launch(void* const* d_in, const int* in_sizes, int n_in,
                              void* d_out, int out_size, void* d_ws, size_t ws_size,
                              hipStream_t stream)
{
    (void)in_sizes; (void)n_in; (void)out_size; (void)d_ws; (void)ws_size;
    const float* pred = (const float*)d_in[0];   // (16, 2048, 5) f32: [score,x,y,w,h]
    int* out = (int*)d_out;                      // (16, 256) int32 indices
    nms_fused_kernel<<<dim3(NMS_B), dim3(256), 0, stream>>>(pred, out);
}